// MultiRelationOntologyGNN_55259049230993
// MI455X (gfx1250) — compile-verified
//
#include <hip/hip_runtime.h>
#include <hip/hip_bf16.h>

// ---------------------------------------------------------------------------
// RGCN (2 layers) for MI455X / gfx1250
//   layer: out = mean-per-(dst,rel) scatter of (X @ W_rel[etype])[src]
//               + X @ W_root + b   (+ ReLU after layer 1)
//
// Strategy: per-relation dense transform into a single 51.2MB buffer that
// stays L2-resident together with agg + activations (192MB L2), so the
// irregular edge gather/scatter hits L2 instead of HBM. GEMMs run on
// V_WMMA_F32_16X16X4_F32 (exact f32 semantics, wave32, 16x16 tile per wave).
// ---------------------------------------------------------------------------

typedef __attribute__((ext_vector_type(2))) float v2f;
typedef __attribute__((ext_vector_type(8))) float v8f;

#define NNODES 100000
#define NEDGES 1600000
#define DIM 128
#define NREL 8

// ---------------------------------------------------------------------------
// H[n, o] = sum_k X[n, k] * W[k, o]        (X: [nrows,128], W: [128,128])
// One wave32 computes a 16x16 tile; 2 waves per block (2 M-tiles).
// grid = (nrows/32, 128/16)
// ---------------------------------------------------------------------------
__global__ __launch_bounds__(64)
void rgcn_transform_wmma(const float* __restrict__ X,
                         const float* __restrict__ W,
                         float* __restrict__ H, int nrows) {
  const int lane  = threadIdx.x & 31;
  const int wave  = threadIdx.x >> 5;
  const int mtile = blockIdx.x * 2 + wave;          // 16-row tile index
  const int ntile = blockIdx.y;                     // 16-col tile index
  if (mtile * 16 >= nrows) return;                  // wave-uniform guard

  const int l15   = lane & 15;
  const int khalf = (lane >> 4) * 2;                // 0 (lanes 0-15) or 2
  const int row   = mtile * 16 + l15;               // A-operand row (M)
  const int col   = ntile * 16 + l15;               // B/C/D column (N)

  const float* __restrict__ xp = X + (size_t)row * DIM;

  v8f acc = {};
  #pragma unroll 8
  for (int k = 0; k < DIM; k += 4) {
    // A 16x4 f32: lane holds A[M=row][K=k+khalf .. k+khalf+1] (contiguous)
    v2f a = *(const v2f*)(xp + k + khalf);
    // B 4x16 f32: lane holds W[K=k+khalf][N=col], W[K=k+khalf+1][N=col]
    v2f b;
    b.x = W[(size_t)(k + khalf)     * DIM + col];
    b.y = W[(size_t)(k + khalf + 1) * DIM + col];
    acc = __builtin_amdgcn_wmma_f32_16x16x4_f32(
        false, a, false, b, (short)0, acc, false, false);
  }

  // C/D layout: acc[i] = D[M = mtile*16 + i + (lane>>4)*8][N = col]
  const int rbase = mtile * 16 + (lane >> 4) * 8;
  #pragma unroll
  for (int i = 0; i < 8; ++i)
    H[(size_t)(rbase + i) * DIM + col] = acc[i];
}

// ---------------------------------------------------------------------------
// out[n,o] = agg[n,o] + (X @ Wroot)[n,o] + b[o]   (+optional ReLU)
// Same tiling as above; accumulator seeded from agg.
// ---------------------------------------------------------------------------
__global__ __launch_bounds__(64)
void rgcn_root_wmma(const float* __restrict__ X,
                    const float* __restrict__ W,
                    const float* __restrict__ bias,
                    const float* __restrict__ agg,
                    float* __restrict__ out, int nrows, int do_relu) {
  const int lane  = threadIdx.x & 31;
  const int wave  = threadIdx.x >> 5;
  const int mtile = blockIdx.x * 2 + wave;
  const int ntile = blockIdx.y;
  if (mtile * 16 >= nrows) return;                  // wave-uniform guard

  const int l15   = lane & 15;
  const int khalf = (lane >> 4) * 2;
  const int row   = mtile * 16 + l15;
  const int col   = ntile * 16 + l15;
  const int rbase = mtile * 16 + (lane >> 4) * 8;

  // Seed C with the aggregated messages.
  v8f acc;
  #pragma unroll
  for (int i = 0; i < 8; ++i)
    acc[i] = agg[(size_t)(rbase + i) * DIM + col];

  const float* __restrict__ xp = X + (size_t)row * DIM;
  #pragma unroll 8
  for (int k = 0; k < DIM; k += 4) {
    v2f a = *(const v2f*)(xp + k + khalf);
    v2f b;
    b.x = W[(size_t)(k + khalf)     * DIM + col];
    b.y = W[(size_t)(k + khalf + 1) * DIM + col];
    acc = __builtin_amdgcn_wmma_f32_16x16x4_f32(
        false, a, false, b, (short)0, acc, false, false);
  }

  const float bv = bias[col];
  #pragma unroll
  for (int i = 0; i < 8; ++i) {
    float v = acc[i] + bv;
    if (do_relu) v = fmaxf(v, 0.0f);
    out[(size_t)(rbase + i) * DIM + col] = v;
  }
}

// ---------------------------------------------------------------------------
// deg[dst*R + etype] += 1  (graph-only, computed once per call)
// ---------------------------------------------------------------------------
__global__ void rgcn_degree(const int* __restrict__ dst,
                            const int* __restrict__ et,
                            float* __restrict__ deg, int nedges) {
  int e = blockIdx.x * blockDim.x + threadIdx.x;
  if (e >= nedges) return;
  atomicAdd(&deg[(size_t)dst[e] * NREL + et[e]], 1.0f);
}

// ---------------------------------------------------------------------------
// For edges of relation r: agg[dst,:] += H[src,:] / max(deg[dst,r],1)
// 32 threads per edge, float4 per thread (128 floats / edge).
// ---------------------------------------------------------------------------
__global__ __launch_bounds__(256)
void rgcn_scatter(const int* __restrict__ src,
                  const int* __restrict__ dst,
                  const int* __restrict__ et,
                  const float* __restrict__ deg,
                  const float* __restrict__ H,
                  float* __restrict__ agg, int nedges, int r) {
  long long t = (long long)blockIdx.x * blockDim.x + threadIdx.x;
  int e = (int)(t >> 5);
  if (e >= nedges) return;
  if (et[e] != r) return;
  const int c = (int)(t & 31) * 4;
  const int s = src[e];
  const int d = dst[e];
  const float norm = 1.0f / fmaxf(deg[(size_t)d * NREL + r], 1.0f);
  const float4 v = *(const float4*)(H + (size_t)s * DIM + c);
  float* ap = agg + (size_t)d * DIM + c;
  atomicAdd(ap + 0, v.x * norm);
  atomicAdd(ap + 1, v.y * norm);
  atomicAdd(ap + 2, v.z * norm);
  atomicAdd(ap + 3, v.w * norm);
}

// ---------------------------------------------------------------------------
__global__ void rgcn_zero(float* __restrict__ p, long long n) {
  long long i = (long long)blockIdx.x * blockDim.x + threadIdx.x;
  long long stride = (long long)gridDim.x * blockDim.x;
  for (; i < n; i += stride) p[i] = 0.0f;
}

// ---------------------------------------------------------------------------
extern "C" void kernel_launch(void* const* d_in, const int* in_sizes, int n_in,
                              void* d_out, int out_size, void* d_ws, size_t ws_size,
                              hipStream_t stream) {
  const float* x      = (const float*)d_in[0];
  const int*   eidx   = (const int*)d_in[1];     // [2, E] flat
  const int*   etype  = (const int*)d_in[2];     // [E]
  const float* Wrel1  = (const float*)d_in[3];   // [R, D, D]
  const float* Wroot1 = (const float*)d_in[4];   // [D, D]
  const float* b1     = (const float*)d_in[5];   // [D]
  const float* Wrel2  = (const float*)d_in[6];
  const float* Wroot2 = (const float*)d_in[7];
  const float* b2     = (const float*)d_in[8];
  float*       out    = (float*)d_out;

  const int* src = eidx;
  const int* dst = eidx + NEDGES;

  // Workspace carve-up (all f32): H | agg | h1 | deg   (~157 MB)
  float* H   = (float*)d_ws;                     // [N, D]  per-relation transform
  float* agg = H   + (size_t)NNODES * DIM;       // [N, D]  aggregated messages
  float* h1  = agg + (size_t)NNODES * DIM;       // [N, D]  layer-1 activations
  float* deg = h1  + (size_t)NNODES * DIM;       // [N, R]  per-(dst,rel) degree

  const dim3 gemmGrid(NNODES / 32, DIM / 16);    // 3125 x 8, 2 waves/block
  const dim3 gemmBlk(64);
  const int  edgeBlks    = (NEDGES + 255) / 256;
  const long long sthr   = (long long)NEDGES * 32;
  const int  scatterBlks = (int)((sthr + 255) / 256);

  // --- degrees (shared by both layers) ---
  rgcn_zero<<<1024, 256, 0, stream>>>(deg, (long long)NNODES * NREL);
  rgcn_degree<<<edgeBlks, 256, 0, stream>>>(dst, etype, deg, NEDGES);

  // ---------------- Layer 1 ----------------
  rgcn_zero<<<4096, 256, 0, stream>>>(agg, (long long)NNODES * DIM);
  for (int r = 0; r < NREL; ++r) {
    rgcn_transform_wmma<<<gemmGrid, gemmBlk, 0, stream>>>(
        x, Wrel1 + (size_t)r * DIM * DIM, H, NNODES);
    rgcn_scatter<<<scatterBlks, 256, 0, stream>>>(
        src, dst, etype, deg, H, agg, NEDGES, r);
  }
  rgcn_root_wmma<<<gemmGrid, gemmBlk, 0, stream>>>(
      x, Wroot1, b1, agg, h1, NNODES, /*relu=*/1);

  // ---------------- Layer 2 ----------------
  rgcn_zero<<<4096, 256, 0, stream>>>(agg, (long long)NNODES * DIM);
  for (int r = 0; r < NREL; ++r) {
    rgcn_transform_wmma<<<gemmGrid, gemmBlk, 0, stream>>>(
        h1, Wrel2 + (size_t)r * DIM * DIM, H, NNODES);
    rgcn_scatter<<<scatterBlks, 256, 0, stream>>>(
        src, dst, etype, deg, H, agg, NEDGES, r);
  }
  rgcn_root_wmma<<<gemmGrid, gemmBlk, 0, stream>>>(
      h1, Wroot2, b2, agg, out, NNODES, /*relu=*/0);
}